// QuantumAutoencoder_65481071400286
// MI455X (gfx1250) — compile-verified
//
#include <hip/hip_runtime.h>

// CDNA5 / gfx1250 wave32 WMMA types
typedef __attribute__((ext_vector_type(2))) float v2f;
typedef __attribute__((ext_vector_type(8))) float v8f;

__device__ __forceinline__ float hsum4(float4 v) {
    return (v.x + v.y) + (v.z + v.w);
}

// One wave32 produces 16 outputs = 2 patch-rows (py0, py0+1) of one image.
// The 12 per-output partial sums (3 channels x 4 rows, each a float4 hsum)
// are reduced with three accumulating V_WMMA_F32_16X16X4_F32 ops:
//   D = A(16x4 partials) x B(4x16 all-ones) + C
// Every column of D equals the row-sum, so the result is layout-proof in B.
__global__ __launch_bounds__(256) void qae_pool_cos_wmma(
    const float* __restrict__ in, float* __restrict__ out, int nWaves)
{
    const int lane = threadIdx.x & 31;
    const int w    = blockIdx.x * 8 + (threadIdx.x >> 5);
    if (w >= nWaves) return;           // wave-uniform: EXEC stays all-ones

    const int b    = w >> 2;           // image index
    const int py0  = (w & 3) << 1;     // first patch-row of this wave: 0,2,4,6
    const int half = lane >> 4;        // A-matrix K-half: 0 -> K=0,1 ; 1 -> K=2,3
    const int m    = lane & 15;        // A-matrix row M = output patch id
    const int py   = py0 + (m >> 3);
    const int px   = m & 7;

    // inputs[b, c, y, x]: c-stride 1024 floats, y-stride 32 floats
    // partial j = c*4 + dy  ->  offset c*1024 + dy*32 from patch base
    const float* p = in + (size_t)b * 3072 + (size_t)py * 128 + (size_t)px * 4;

    v8f acc = {0.f, 0.f, 0.f, 0.f, 0.f, 0.f, 0.f, 0.f};
    v2f bones; bones[0] = 1.0f; bones[1] = 1.0f;

    #pragma unroll
    for (int k0 = 0; k0 < 12; k0 += 4) {
        const int j0 = k0 + 2 * half;
        const int j1 = j0 + 1;
        const float4 q0 = *(const float4*)(p + (j0 >> 2) * 1024 + (j0 & 3) * 32);
        const float4 q1 = *(const float4*)(p + (j1 >> 2) * 1024 + (j1 & 3) * 32);
        v2f a; a[0] = hsum4(q0); a[1] = hsum4(q1);
        // 8-arg form: (neg_a, A, neg_b, B, c_mod, C, reuse_a, reuse_b)
        acc = __builtin_amdgcn_wmma_f32_16x16x4_f32(
            false, a, false, bones, (short)0, acc, false, false);
    }

    // D layout (16x16 f32): lanes 0-15, VGPR v -> M=v; lanes 16-31 -> M=v+8.
    // All columns identical, so lane 0 holds outputs 0..7, lane 16 holds 8..15.
    if (m == 0) {
        const float kScale  = 3.14159265358979323846f / (48.0f * 255.0f);
        const float kHalfPi = 1.57079632679489661923f;
        const float kInv2Pi = 0.15915494309189535f;   // v_cos_f32 takes revolutions
        float r[8];
        #pragma unroll
        for (int v = 0; v < 8; ++v) {
            float ang = acc[v] * kScale - kHalfPi;
            r[v] = __builtin_amdgcn_cosf(ang * kInv2Pi);
        }
        float* o = out + (size_t)b * 64 + (size_t)py0 * 8 + (size_t)half * 8;
        *(float4*)(o)     = make_float4(r[0], r[1], r[2], r[3]);
        *(float4*)(o + 4) = make_float4(r[4], r[5], r[6], r[7]);
    }
}

extern "C" void kernel_launch(void* const* d_in, const int* in_sizes, int n_in,
                              void* d_out, int out_size, void* d_ws, size_t ws_size,
                              hipStream_t stream) {
    const float* in = (const float*)d_in[0];
    float* out      = (float*)d_out;

    const int B      = in_sizes[0] / 3072;   // 3*32*32 floats per image
    const int nWaves = B * 4;                // 4 waves (16 outputs each) per image
    const int blocks = (nWaves + 7) / 8;     // 8 wave32s per 256-thread block

    qae_pool_cos_wmma<<<blocks, 256, 0, stream>>>(in, out, nWaves);
}